// ChargeEq_17420387352800
// MI455X (gfx1250) — compile-verified
//
#include <hip/hip_runtime.h>
#include <hip/hip_bf16.h>
#include <math.h>

typedef __attribute__((ext_vector_type(2))) float v2f;
typedef __attribute__((ext_vector_type(8))) float v8f;

#define B_ 8
#define N_ 256
#define NMAX_ 10
#define GRID1 21
#define GRIDTOT (GRID1*GRID1*GRID1)   /* 9261 */
#define KVEC_STRIDE 4640              /* max compacted half-grid k's (<=4633 used) */
#define KP2 9280                      /* P row stride in floats (2*KVEC_STRIDE)   */
#define TWO_PI_F 6.28318530717958647692f
#define KSQ_MAX_F ((TWO_PI_F/1.5f)*(TWO_PI_F/1.5f))
#define DIAG_F 0.79788456080286535588f  /* 2/sqrt(2*pi), SIGMA=1 */

/* workspace byte offsets */
#define HDR_OFF_B   0
#define KVEC_OFF_B  256
#define P_OFF_B     (KVEC_OFF_B + (size_t)B_*KVEC_STRIDE*16)
#define A_OFF_B     (P_OFF_B + (size_t)B_*N_*KP2*4)

/* ---- kernel 0: per-batch k-vector compaction (deterministic ballot scan) ---- */
__global__ void kvec_build(const float* __restrict__ cell, int* __restrict__ hdr,
                           float4* __restrict__ kvecs)
{
    int b = blockIdx.x;
    int lane = threadIdx.x;          /* block = 32 threads = one wave32 */
    const float* C = cell + b*9;
    float c00=C[0],c01=C[1],c02=C[2],c10=C[3],c11=C[4],c12=C[5],c20=C[6],c21=C[7],c22=C[8];
    float det = c00*(c11*c22-c12*c21) - c01*(c10*c22-c12*c20) + c02*(c10*c21-c11*c20);
    float id = 1.0f/det;
    float i00 =  (c11*c22-c12*c21)*id, i01 = -(c01*c22-c02*c21)*id, i02 =  (c01*c12-c02*c11)*id;
    float i10 = -(c10*c22-c12*c20)*id, i11 =  (c00*c22-c02*c20)*id, i12 = -(c00*c12-c02*c10)*id;
    float i20 =  (c10*c21-c11*c20)*id, i21 = -(c00*c21-c01*c20)*id, i22 =  (c00*c11-c01*c10)*id;
    /* recip row r = 2*pi * column r of inv(cell) */
    float r0x = TWO_PI_F*i00, r0y = TWO_PI_F*i10, r0z = TWO_PI_F*i20;
    float r1x = TWO_PI_F*i01, r1y = TWO_PI_F*i11, r1z = TWO_PI_F*i21;
    float r2x = TWO_PI_F*i02, r2y = TWO_PI_F*i12, r2z = TWO_PI_F*i22;
    float vol = fabsf(det);
    float pref = 2.0f * TWO_PI_F / vol * 2.0f;   /* extra x2: half-grid folding */

    float4* kv = kvecs + (size_t)b*KVEC_STRIDE;
    unsigned base = 0;
    for (int start = 0; start < GRIDTOT; start += 32) {
        int idx = start + lane;
        bool pred = false;
        float kx=0.f,ky=0.f,kz=0.f,g=0.f;
        if (idx < GRIDTOT) {
            int nx = idx/(GRID1*GRID1) - NMAX_;
            int ny = (idx/GRID1)%GRID1 - NMAX_;
            int nz = idx%GRID1        - NMAX_;
            bool hemi = (nz>0) || (nz==0 && ny>0) || (nz==0 && ny==0 && nx>0);
            if (hemi) {
                float fnx=(float)nx, fny=(float)ny, fnz=(float)nz;
                kx = fnx*r0x + fny*r1x + fnz*r2x;
                ky = fnx*r0y + fny*r1y + fnz*r2y;
                kz = fnx*r0z + fny*r1z + fnz*r2z;
                float k2 = kx*kx + ky*ky + kz*kz;
                if (k2 > 1e-10f && k2 < KSQ_MAX_F) {
                    float f = expf(-0.5f*k2)/k2;
                    g = sqrtf(pref*f);
                    pred = true;
                }
            }
        }
        unsigned mask = (unsigned)__ballot(pred);
        unsigned off = __popc(mask & ((1u<<lane)-1u));
        if (pred) kv[base + off] = make_float4(kx,ky,kz,g);
        base += __popc(mask);
    }
    int Kv = (int)base;
    int KvPad = (Kv + 3) & ~3;       /* columns = 2*KvPad -> multiple of 8 */
    if (lane < 4 && (Kv + lane) < KVEC_STRIDE)
        kv[Kv + lane] = make_float4(0.f,0.f,0.f,0.f);
    if (lane == 0) { hdr[b*2] = Kv; hdr[b*2+1] = KvPad; }
}

/* ---- kernel 1: P[i][2k]=g*cos(k.r), P[i][2k+1]=g*sin(k.r) ---- */
__global__ void phase_build(const float* __restrict__ pos, const int* __restrict__ hdr,
                            const float4* __restrict__ kvecs, float* __restrict__ P)
{
    int i = blockIdx.x, b = blockIdx.y;
    int row = b*N_ + i;
    float rx = pos[row*3+0], ry = pos[row*3+1], rz = pos[row*3+2];
    int KvPad = hdr[b*2+1];
    const float4* kv = kvecs + (size_t)b*KVEC_STRIDE;
    float2* Prow = (float2*)(P + (size_t)row*KP2);
    for (int kk = threadIdx.x; kk < KvPad; kk += blockDim.x) {
        float4 k4 = kv[kk];
        float ph = k4.x*rx + k4.y*ry + k4.z*rz;
        float s, c;
        __sincosf(ph, &s, &c);
        Prow[kk] = make_float2(k4.w*c, k4.w*s);
    }
}

/* ---- kernel 2: A = P * P^T via V_WMMA_F32_16X16X4_F32, upper tiles + mirror ---- */
__global__ void __launch_bounds__(128) ewald_gemm(const float* __restrict__ P,
                                                  const int* __restrict__ hdr,
                                                  float* __restrict__ A)
{
    int b    = blockIdx.z;
    int wave = threadIdx.x >> 5;
    int lane = threadIdx.x & 31;
    int tlin = blockIdx.x*4 + wave;      /* 0..135 upper-triangle tiles */
    int t = tlin, ti = 0, rem = 16;
    while (t >= rem) { t -= rem; ti++; rem--; }
    int tj = ti + t;

    int cols = 2*hdr[b*2+1];             /* multiple of 8 */
    int m16 = lane & 15, hf = lane >> 4;
    const float* Pb   = P + (size_t)b*N_*KP2;
    const float* arow = Pb + (size_t)(ti*16 + m16)*KP2 + 2*hf;
    const float* brow = Pb + (size_t)(tj*16 + m16)*KP2 + 2*hf;

    v8f acc0 = {0.f,0.f,0.f,0.f,0.f,0.f,0.f,0.f};
    v8f acc1 = {0.f,0.f,0.f,0.f,0.f,0.f,0.f,0.f};
    for (int k0 = 0; k0 < cols; k0 += 8) {
        v2f a0 = *(const v2f*)(arow + k0);
        v2f b0 = *(const v2f*)(brow + k0);
        v2f a1 = *(const v2f*)(arow + k0 + 4);
        v2f b1 = *(const v2f*)(brow + k0 + 4);
        acc0 = __builtin_amdgcn_wmma_f32_16x16x4_f32(false, a0, false, b0,
                                                     (short)0, acc0, false, false);
        acc1 = __builtin_amdgcn_wmma_f32_16x16x4_f32(false, a1, false, b1,
                                                     (short)0, acc1, false, false);
    }
    v8f acc = acc0 + acc1;

    float* Ab = A + (size_t)b*N_*N_;
    int j = tj*16 + m16;
#pragma unroll
    for (int v = 0; v < 8; ++v) {
        int i = ti*16 + v + 8*hf;
        float val = acc[v];
        Ab[(size_t)i*N_ + j] = val - ((i==j) ? DIAG_F : 0.0f);
        if (ti != tj) Ab[(size_t)j*N_ + i] = val;
    }
}

/* ---- kernel 3: bordered 257x257 LU solve in 265KB LDS + energy ---- */
__global__ void __launch_bounds__(256) qeq_solve(const float* __restrict__ A,
                                                 const float* __restrict__ chi,
                                                 const float* __restrict__ Jraw,
                                                 const int*   __restrict__ zat,
                                                 const float* __restrict__ Q,
                                                 float* __restrict__ out)
{
    extern __shared__ float M[];             /* 257 x 258 floats = 265,224 B */
    __shared__ float red[256];
    __shared__ int   redi[256];
    __shared__ float xs[N_+1];
    const int n = N_ + 1, ldm = N_ + 2;
    int b = blockIdx.x, tid = threadIdx.x;
    const float* Ab = A + (size_t)b*N_*N_;

    /* assemble augmented matrix [A+diag(J), 1 | -chi ; 1^T, 0 | Q/NORM] */
    for (int e = tid; e < n*ldm; e += 256) {
        int r = e / ldm, c = e - r*ldm;
        float v;
        if (c == N_ + 1) {
            v = (r < N_) ? -chi[b*N_ + r] : Q[b] * sqrtf(90.0474f);
        } else if (r < N_ && c < N_) {
            v = Ab[(size_t)r*N_ + c];
            if (r == c) {
                int z = zat[b*N_ + r];
                int zi = (z==1) ? 0 : (z==6) ? 1 : (z==7) ? 2 : 3;
                float jr = Jraw[zi];
                v += jr*jr;
            }
        } else if (r == N_ && c == N_) {
            v = 0.0f;
        } else {
            v = 1.0f;
        }
        M[r*ldm + c] = v;
    }
    __syncthreads();

    /* forward elimination with partial pivoting (rhs in col N_+1) */
    for (int p = 0; p < n-1; ++p) {
        float bv = -1.0f; int bi = p;
        for (int r = p + tid; r < n; r += 256) {
            float av = fabsf(M[r*ldm + p]);
            if (av > bv) { bv = av; bi = r; }
        }
        red[tid] = bv; redi[tid] = bi;
        __syncthreads();
        for (int s = 128; s > 0; s >>= 1) {
            if (tid < s) {
                if (red[tid+s] > red[tid] ||
                    (red[tid+s] == red[tid] && redi[tid+s] < redi[tid])) {
                    red[tid] = red[tid+s]; redi[tid] = redi[tid+s];
                }
            }
            __syncthreads();
        }
        int piv = redi[0];
        if (piv != p) {
            for (int c = p + tid; c < ldm; c += 256) {
                float tv = M[p*ldm + c];
                M[p*ldm + c] = M[piv*ldm + c];
                M[piv*ldm + c] = tv;
            }
        }
        __syncthreads();
        float invp = 1.0f / M[p*ldm + p];
        int r = p + 1 + tid;
        if (r < n) {
            float f = M[r*ldm + p] * invp;
            const float* prow = &M[p*ldm];
            float* rrow = &M[r*ldm];
            for (int c = p + 1; c < ldm; ++c) rrow[c] -= f * prow[c];
        }
        __syncthreads();
    }

    /* back substitution */
    for (int p = n-1; p >= 0; --p) {
        float part = 0.0f;
        for (int c = p + 1 + tid; c < n; c += 256) part += M[p*ldm + c] * xs[c];
        red[tid] = part;
        __syncthreads();
        for (int s = 128; s > 0; s >>= 1) {
            if (tid < s) red[tid] += red[tid+s];
            __syncthreads();
        }
        if (tid == 0) xs[p] = (M[p*ldm + (N_+1)] - red[0]) / M[p*ldm + p];
        __syncthreads();
    }

    /* q output */
    out[b*N_ + tid] = xs[tid];

    /* e = 0.5 * q^T A q  (A from workspace, L2-resident) */
    float qi = xs[tid];
    const float* arow2 = Ab + (size_t)tid*N_;
    float dot = 0.0f;
    for (int j2 = 0; j2 < N_; ++j2) dot += arow2[j2] * xs[j2];
    red[tid] = qi * dot;
    __syncthreads();
    for (int s = 128; s > 0; s >>= 1) {
        if (tid < s) red[tid] += red[tid+s];
        __syncthreads();
    }
    if (tid == 0) out[B_*N_ + b] = 0.5f * red[0];
}

extern "C" void kernel_launch(void* const* d_in, const int* in_sizes, int n_in,
                              void* d_out, int out_size, void* d_ws, size_t ws_size,
                              hipStream_t stream) {
    const float* pos  = (const float*)d_in[0];
    const float* cell = (const float*)d_in[1];
    const float* chi  = (const float*)d_in[2];
    const float* Jraw = (const float*)d_in[3];
    const float* Q    = (const float*)d_in[4];
    const int*   zat  = (const int*)d_in[5];
    float* out = (float*)d_out;

    char* ws = (char*)d_ws;
    int*    hdr   = (int*)(ws + HDR_OFF_B);
    float4* kvecs = (float4*)(ws + KVEC_OFF_B);
    float*  P     = (float*)(ws + P_OFF_B);
    float*  A     = (float*)(ws + A_OFF_B);

    hipLaunchKernelGGL(kvec_build, dim3(B_), dim3(32), 0, stream, cell, hdr, kvecs);
    hipLaunchKernelGGL(phase_build, dim3(N_, B_), dim3(256), 0, stream, pos, hdr, kvecs, P);
    hipLaunchKernelGGL(ewald_gemm, dim3(34, 1, B_), dim3(128), 0, stream, P, hdr, A);

    size_t lds = (size_t)(N_+1)*(N_+2)*sizeof(float);  /* 265,224 B < 320KB WGP LDS */
    hipFuncSetAttribute(reinterpret_cast<const void*>(qeq_solve),
                        hipFuncAttributeMaxDynamicSharedMemorySize, (int)lds);
    hipLaunchKernelGGL(qeq_solve, dim3(B_), dim3(256), lds, stream,
                       A, chi, Jraw, zat, Q, out);
}